// KDEGaussian_76665166233781
// MI455X (gfx1250) — compile-verified
//
#include <hip/hip_runtime.h>

typedef __attribute__((ext_vector_type(2))) float v2f;
typedef __attribute__((ext_vector_type(8))) float v8f;

#define LOG2E 1.44269504088896340736f
#define NCHUNK_PER_B 4
#define NCHUNKS 8   // NCHUNK_PER_B * B(=2)

// ---------------------------------------------------------------------------
// Main kernel: each wave owns one 16-location tile and one sample chunk.
// A = samples (M x K=4), B = locations (K=4 x N=16); one v_wmma_f32_16x16x4_f32
// yields the full exp2-argument tile: s*dist^2 with s = -1/(2*bw^2) * log2(e).
// ---------------------------------------------------------------------------
__global__ __launch_bounds__(256) void kde_wmma_kernel(
    const float* __restrict__ samples,    // (B*S, 2) flat
    const float* __restrict__ locations,  // (m, 2) flat
    float* __restrict__ partials,         // (NCHUNKS, m)
    int m, int S, int rowsPerChunk)
{
    const int wid  = (blockIdx.x * blockDim.x + threadIdx.x) >> 5;
    const int lane = threadIdx.x & 31;
    const int nIdx = lane & 15;
    const int hi   = lane >> 4;

    const int loc_tile = wid / NCHUNKS;
    const int chunk    = wid % NCHUNKS;
    const int b        = chunk >> 2;            // which batch (B=2)
    const int c        = chunk & (NCHUNK_PER_B - 1);

    // -------- B matrix: locations (loop invariant) --------
    // lane L: N = L%16 ; v0/v1 hold K = 2*(L/16)+0/1
    const float2* loc2 = reinterpret_cast<const float2*>(locations);
    float2 y = loc2[loc_tile * 16 + nIdx];
    v2f bm;
    if (hi) { bm[0] = 1.0f;       bm[1] = y.x * y.x + y.y * y.y; }
    else    { bm[0] = -2.0f * y.x; bm[1] = -2.0f * y.y; }

    // fold 1/(2*bw^2)=12.5 and ln->log2 into the A matrix
    const float s = -12.5f * LOG2E;

    const float2* smp2 = reinterpret_cast<const float2*>(samples);
    const int rowBase = b * S + c * rowsPerChunk;
    const int nTiles  = rowsPerChunk >> 4;

    v8f acc = {0.f, 0.f, 0.f, 0.f, 0.f, 0.f, 0.f, 0.f};

    for (int t = 0; t < nTiles; ++t) {
        // -------- A matrix: 16 sample rows of this tile --------
        int row = rowBase + t * 16 + nIdx;
        float2 x = smp2[row];
        v2f am;
        if (hi) { am[0] = s * (x.x * x.x + x.y * x.y); am[1] = s; }
        else    { am[0] = s * x.x;                     am[1] = s * x.y; }

        v8f cz = {0.f, 0.f, 0.f, 0.f, 0.f, 0.f, 0.f, 0.f};
        // D[M=sample, N=location] = s * dist^2  (exp2 argument)
        v8f d = __builtin_amdgcn_wmma_f32_16x16x4_f32(
            false, am, false, bm, (short)0, cz, false, false);

#pragma unroll
        for (int i = 0; i < 8; ++i)
            acc[i] += __builtin_amdgcn_exp2f(d[i]);
    }

    // in-lane fold over the 8 sample-row accumulators
    float tsum = ((acc[0] + acc[1]) + (acc[2] + acc[3])) +
                 ((acc[4] + acc[5]) + (acc[6] + acc[7]));
    // fold the two M-halves (lanes L and L+16 share the same location N)
    tsum += __shfl_xor(tsum, 16);

    if (hi == 0)
        partials[chunk * m + loc_tile * 16 + nIdx] = tsum;
}

// ---------------------------------------------------------------------------
// Reduce the NCHUNK_PER_B partials per (j,b) into out[j*2+b], and compute
// norm[b] with a deterministic LDS tree reduction (one block per b).
// ---------------------------------------------------------------------------
__global__ __launch_bounds__(256) void kde_reduce_kernel(
    const float* __restrict__ partials, float* __restrict__ out,
    float* __restrict__ norms, int m)
{
    const int b = blockIdx.x;
    const int tid = threadIdx.x;

    float lnorm = 0.0f;
    for (int j = tid; j < m; j += blockDim.x) {
        float o = (partials[(b * NCHUNK_PER_B + 0) * m + j] +
                   partials[(b * NCHUNK_PER_B + 1) * m + j]) +
                  (partials[(b * NCHUNK_PER_B + 2) * m + j] +
                   partials[(b * NCHUNK_PER_B + 3) * m + j]);
        out[j * 2 + b] = o;
        lnorm += o;
    }

    __shared__ float sd[256];
    sd[tid] = lnorm;
    __syncthreads();
    for (int stp = 128; stp > 0; stp >>= 1) {
        if (tid < stp) sd[tid] += sd[tid + stp];
        __syncthreads();
    }
    if (tid == 0) norms[b] = sd[0];
}

// ---------------------------------------------------------------------------
// pdf[j*2+b] = out[j*2+b] / norm[b]
// ---------------------------------------------------------------------------
__global__ __launch_bounds__(256) void kde_norm_kernel(
    float* __restrict__ out, const float* __restrict__ norms, int n)
{
    int i = blockIdx.x * blockDim.x + threadIdx.x;
    if (i < n) out[i] = out[i] / norms[i & 1];
}

extern "C" void kernel_launch(void* const* d_in, const int* in_sizes, int n_in,
                              void* d_out, int out_size, void* d_ws, size_t ws_size,
                              hipStream_t stream) {
    const float* samples   = (const float*)d_in[0];  // (B, S, 2) f32
    const float* locations = (const float*)d_in[1];  // (m, 2)    f32
    float* out = (float*)d_out;

    const int R = in_sizes[0] / 2;   // B*S = 8192
    const int m = in_sizes[1] / 2;   // 8192
    const int B = 2;
    const int S = R / B;             // 4096
    const int rowsPerChunk = S / NCHUNK_PER_B;  // 1024

    float* partials = (float*)d_ws;                    // NCHUNKS * m floats
    float* norms    = partials + (size_t)NCHUNKS * m;  // 2 floats

    // one wave (32 threads) per (location-tile, chunk) pair
    const int nWaves  = (m / 16) * NCHUNKS;            // 4096
    const int threads = 256;
    const int blocks  = nWaves * 32 / threads;         // 512

    kde_wmma_kernel<<<blocks, threads, 0, stream>>>(
        samples, locations, partials, m, S, rowsPerChunk);
    kde_reduce_kernel<<<B, 256, 0, stream>>>(partials, out, norms, m);

    const int n = 2 * m;
    kde_norm_kernel<<<(n + 255) / 256, 256, 0, stream>>>(out, norms, n);
}